// ContrastiveLoss_rec_31550829756779
// MI455X (gfx1250) — compile-verified
//
#include <hip/hip_runtime.h>
#include <hip/hip_bf16.h>

typedef float v2f __attribute__((ext_vector_type(2)));
typedef float v8f __attribute__((ext_vector_type(8)));

#define MARGIN 0.2f
#define E_DIM 1024
#define BM 128
#define BN 64
#define BK 32
#define US_STRIDE (BK + 4)    /* 36 words: conflict-free strided A-frag reads */
#define WS2_STRIDE (BN + 16)  /* 80 float2 = 160 words == 32 mod 64: halves on disjoint banks */

__global__ void __launch_bounds__(256) zero_f32_kernel(float* __restrict__ p, int n) {
  int i = blockIdx.x * blockDim.x + threadIdx.x;
  if (i < n) p[i] = 0.0f;
}

// P = U @ W with U = 0.4*m + 0.6*tr_m computed on the fly (reduction over e,
// W indexed [e, n] row-major => coalesced B loads), then rowwise dot with
// D = A_is - A_em, per-row partials accumulated atomically into delta[].
__global__ void __launch_bounds__(256) fused_gemm_rowdot_kernel(
    const float* __restrict__ Ais, const float* __restrict__ Aem,
    const float* __restrict__ mp,  const float* __restrict__ trm,
    const float* __restrict__ W,   float* __restrict__ delta) {
  __shared__ float Us[BM * US_STRIDE];
  // W tile in K-pair-interleaved layout: Ws2[kp][n] = {W[2kp][n], W[2kp+1][n]}
  // so a B-fragment is ONE contiguous 8-byte LDS read (ds_load_b64), no movs.
  __shared__ v2f Ws2[(BK / 2) * WS2_STRIDE];

  const int tid  = threadIdx.x;
  const int wave = tid >> 5;
  const int lane = tid & 31;
  const int hsel = lane >> 4;   // which 16-lane half of the wave
  const int l15  = lane & 15;

  const int n0 = blockIdx.x * BN;
  const int m0 = blockIdx.y * BM;

  v8f acc[4] = {v8f{}, v8f{}, v8f{}, v8f{}};

  for (int k0 = 0; k0 < E_DIM; k0 += BK) {
    __syncthreads();
    // Stage U tile [BM x BK], fusing u = 0.4*m + 0.6*tr_m. 1024 float4 slots.
    #pragma unroll
    for (int s = tid; s < BM * (BK / 4); s += 256) {
      int row = s >> 3;            // BK/4 == 8 slots per row
      int c4  = (s & 7) << 2;
      int g   = (m0 + row) * E_DIM + k0 + c4;
      float4 a = *(const float4*)(mp  + g);
      float4 b = *(const float4*)(trm + g);
      float* dst = &Us[row * US_STRIDE + c4];
      dst[0] = 0.4f * a.x + 0.6f * b.x;
      dst[1] = 0.4f * a.y + 0.6f * b.y;
      dst[2] = 0.4f * a.z + 0.6f * b.z;
      dst[3] = 0.4f * a.w + 0.6f * b.w;
    }
    // Stage W tile [BK x BN] (rows e = k0.., cols n0..) into pair layout.
    float* wsf = (float*)Ws2;
    #pragma unroll
    for (int s = tid; s < BK * (BN / 4); s += 256) {
      int row  = s >> 4;           // 0..31
      int c4   = (s & 15) << 2;
      int kp   = row >> 1;
      int comp = row & 1;
      int g    = (k0 + row) * E_DIM + n0 + c4;
      float4 w = *(const float4*)(W + g);
      float* dst = wsf + (kp * WS2_STRIDE + c4) * 2 + comp;
      dst[0] = w.x; dst[2] = w.y; dst[4] = w.z; dst[6] = w.w;
    }
    // Prefetch next K-chunk into cache while this chunk computes.
    if (k0 + BK < E_DIM) {
      const int kn = k0 + BK;
      if (tid < BM) {
        __builtin_prefetch(mp  + (m0 + tid) * E_DIM + kn, 0, 1);
        __builtin_prefetch(trm + (m0 + tid) * E_DIM + kn, 0, 1);
      } else if (tid < BM + BK) {
        __builtin_prefetch(W + (kn + tid - BM) * E_DIM + n0, 0, 1);
      }
    }
    __syncthreads();

    const int mrow = wave * 16 + l15;   // A-frag: both lane halves hold M=l15
    #pragma unroll
    for (int kk = 0; kk < BK; kk += 4) {
      // A-frag (16x4 f32): VGPR0 = K={0|2}, VGPR1 = K={1|3} by lane half
      v2f a = *(const v2f*)(&Us[mrow * US_STRIDE + kk + 2 * hsel]);
      // B pair-row for this k-step and lane half
      const v2f* brow = &Ws2[((kk >> 1) + hsel) * WS2_STRIDE + l15];
      #pragma unroll
      for (int t = 0; t < 4; ++t) {
        v2f b = brow[t * 16];      // single ds_load_b64, even-aligned pair
        acc[t] = __builtin_amdgcn_wmma_f32_16x16x4_f32(
            false, a, false, b, (short)0, acc[t], false, false);
      }
    }
  }

  // Epilogue: C layout — lane l, VGPR v holds (M = v + 8*(l>>4), N = l&15).
  // Dot each P element with D = A_is - A_em, reduce N across the 16-lane half,
  // atomically accumulate per-row partial deltas.
  float sv[8];
  #pragma unroll
  for (int v = 0; v < 8; ++v) sv[v] = 0.0f;
  #pragma unroll
  for (int t = 0; t < 4; ++t) {
    int col = n0 + t * 16 + l15;
    #pragma unroll
    for (int v = 0; v < 8; ++v) {
      int row = m0 + wave * 16 + hsel * 8 + v;
      int g = row * E_DIM + col;
      sv[v] += acc[t][v] * (Ais[g] - Aem[g]);
    }
  }
  #pragma unroll
  for (int v = 0; v < 8; ++v) {
    float s = sv[v];
    s += __shfl_xor(s, 1, 32);
    s += __shfl_xor(s, 2, 32);
    s += __shfl_xor(s, 4, 32);
    s += __shfl_xor(s, 8, 32);
    if (l15 == 0) {
      atomicAdd(&delta[m0 + wave * 16 + hsel * 8 + v], s);
    }
  }
}

__global__ void __launch_bounds__(256) hinge_reduce_kernel(
    const float* __restrict__ delta, float* __restrict__ out, int B) {
  __shared__ float red[256];
  float s = 0.0f;
  for (int r = threadIdx.x; r < B; r += 256) {
    s += fmaxf(MARGIN + delta[r], 0.0f);
  }
  red[threadIdx.x] = s;
  __syncthreads();
  for (int o = 128; o > 0; o >>= 1) {
    if (threadIdx.x < (unsigned)o) red[threadIdx.x] += red[threadIdx.x + o];
    __syncthreads();
  }
  if (threadIdx.x == 0) out[0] = red[0];
}

extern "C" void kernel_launch(void* const* d_in, const int* in_sizes, int n_in,
                              void* d_out, int out_size, void* d_ws, size_t ws_size,
                              hipStream_t stream) {
  const float* Ais = (const float*)d_in[0];
  const float* Aem = (const float*)d_in[1];
  const float* m   = (const float*)d_in[2];
  const float* trm = (const float*)d_in[3];
  const float* W   = (const float*)d_in[4];
  // d_in[5] = b: cancels exactly in (diag_is - diag_em), not needed.
  (void)n_in; (void)out_size; (void)ws_size;

  const int B = in_sizes[0] / E_DIM;   // 8192
  float* delta = (float*)d_ws;          // B floats of scratch

  zero_f32_kernel<<<(B + 255) / 256, 256, 0, stream>>>(delta, B);

  dim3 grid(E_DIM / BN, B / BM);        // (16, 64)
  fused_gemm_rowdot_kernel<<<grid, 256, 0, stream>>>(Ais, Aem, m, trm, W, delta);

  hinge_reduce_kernel<<<1, 256, 0, stream>>>(delta, (float*)d_out, B);
}